// Sampler_8658654069314
// MI455X (gfx1250) — compile-verified
//
#include <hip/hip_runtime.h>
#include <stdint.h>

// Temperature sampler: B=256 rows, V=128000 fp32 logits.
// out[row] = temp==0 ? argmax(logits[row]) : gumbel-max(logits[row]/temp).
//
// Pure HBM-bandwidth bound: 131 MB @ 23.3 TB/s ~= 5.6 us floor.
// BDP = 23.3 TB/s * ~600 ns ~= 14 MB must be in flight, so:
//   - 5 segments/row  -> 1280 workgroups (wide WGP coverage)
//   - 8-deep async-to-LDS pipeline (GLOBAL_LOAD_ASYNC_TO_LDS_B128 + ASYNCcnt),
//     7 chunks outstanding per wave -> ~37 MB issued window.
// Stage 1 writes per-segment (argmax, gumbel-argmax) partials to d_ws;
// stage 2 (256 threads) merges segments in order and applies the temp==0 select.

#define VDIM    128000
#define SEGS    5
#define SEGV    (VDIM / SEGS)        // 25600 floats per block
#define BLOCK   256
#define CHUNK   1024                 // floats per chunk per block (256 thr * 4)
#define NCHUNK  (SEGV / CHUNK)       // 25 (exact)
#define NBUF    8                    // 32 KB LDS staging, 7 chunks in flight

__device__ __forceinline__ uint32_t hashu(uint32_t x) {
    x ^= 42u * 0x9E3779B9u;          // seed 42 mix
    x ^= x >> 16; x *= 0x7FEB352Du;
    x ^= x >> 15; x *= 0x846CA68Bu;
    x ^= x >> 16;
    return x;
}

__device__ __forceinline__ float gumbel_noise(uint32_t idx) {
    uint32_t h = hashu(idx);
    float u = (float)(h >> 8) * 0x1p-24f + 0x1p-25f;   // uniform in (0,1)
    return -__logf(-__logf(u));
}

// ---------------- stage 1: per-segment streaming reduction ----------------
__global__ __launch_bounds__(BLOCK) void sampler_partial_kernel(
    const float* __restrict__ logits,
    const float* __restrict__ temps,
    float4* __restrict__ partial)    // [B*SEGS] {gV, asint(gI), sV, asint(sI)}
{
    __shared__ float stage[NBUF * CHUNK];   // 32 KB async staging
    __shared__ float redGV[8]; __shared__ int redGI[8];
    __shared__ float redSV[8]; __shared__ int redSI[8];

    const int row  = blockIdx.x / SEGS;
    const int seg  = blockIdx.x % SEGS;
    const int tid  = threadIdx.x;
    const int base = seg * SEGV;                    // absolute column base
    const float* segp = logits + (size_t)row * VDIM + base;

    const float t    = temps[row];
    const float invT = (t == 0.0f) ? 1.0f : (1.0f / t);

    // Flat LDS aperture: low 32 bits of a generic LDS pointer == LDS byte addr.
    const uint32_t ldsBase = (uint32_t)(uintptr_t)(&stage[0]);

    auto issue = [&](int c) {   // each lane DMAs its own 16 B global -> LDS
        uint64_t ga  = (uint64_t)(segp + c * CHUNK + tid * 4);
        uint32_t dst = ldsBase + (uint32_t)(((c & (NBUF - 1)) * CHUNK + tid * 4) * 4);
        asm volatile("global_load_async_to_lds_b128 %0, %1, off"
                     :: "v"(dst), "v"(ga) : "memory");
    };

    float gV = -3.4e38f; int gI = 0;     // greedy argmax (raw logits)
    float sV = -3.4e38f; int sI = 0;     // gumbel-max (scaled logits)

    auto process = [&](int c) {
        const float4 v = *reinterpret_cast<const float4*>(
            &stage[(c & (NBUF - 1)) * CHUNK + tid * 4]);
        const int col0 = base + c * CHUNK + tid * 4;   // absolute column
        float vals[4] = { v.x, v.y, v.z, v.w };
#pragma unroll
        for (int j = 0; j < 4; ++j) {
            const float x  = vals[j];
            const int  col = col0 + j;
            if (x > gV) { gV = x; gI = col; }          // strict > keeps first idx
            const float key = x * invT +
                gumbel_noise((uint32_t)row * (uint32_t)VDIM + (uint32_t)col);
            if (key > sV) { sV = key; sI = col; }
        }
    };

    // ---- 8-deep async pipeline; async loads retire in order; each lane only
    // reads bytes it DMA'd itself -> no barriers in the steady-state loop ----
    issue(0); issue(1); issue(2); issue(3); issue(4); issue(5); issue(6);
    for (int c = 0; c < NCHUNK - 7; ++c) {
        issue(c + 7);
        asm volatile("s_wait_asynccnt 0x7" ::: "memory");  // chunk c complete
        process(c);
    }
    asm volatile("s_wait_asynccnt 0x6" ::: "memory"); process(NCHUNK - 7);
    asm volatile("s_wait_asynccnt 0x5" ::: "memory"); process(NCHUNK - 6);
    asm volatile("s_wait_asynccnt 0x4" ::: "memory"); process(NCHUNK - 5);
    asm volatile("s_wait_asynccnt 0x3" ::: "memory"); process(NCHUNK - 4);
    asm volatile("s_wait_asynccnt 0x2" ::: "memory"); process(NCHUNK - 3);
    asm volatile("s_wait_asynccnt 0x1" ::: "memory"); process(NCHUNK - 2);
    asm volatile("s_wait_asynccnt 0x0" ::: "memory"); process(NCHUNK - 1);

    // ---- wave32 reduction (first-index tie-break) ----
#pragma unroll
    for (int off = 16; off > 0; off >>= 1) {
        float oGV = __shfl_xor(gV, off, 32); int oGI = __shfl_xor(gI, off, 32);
        if (oGV > gV || (oGV == gV && oGI < gI)) { gV = oGV; gI = oGI; }
        float oSV = __shfl_xor(sV, off, 32); int oSI = __shfl_xor(sI, off, 32);
        if (oSV > sV || (oSV == sV && oSI < sI)) { sV = oSV; sI = oSI; }
    }

    const int wave = tid >> 5;
    if ((tid & 31) == 0) {
        redGV[wave] = gV; redGI[wave] = gI;
        redSV[wave] = sV; redSI[wave] = sI;
    }
    __syncthreads();

    if (tid == 0) {
        float bGV = redGV[0]; int bGI = redGI[0];
        float bSV = redSV[0]; int bSI = redSI[0];
#pragma unroll
        for (int w = 1; w < 8; ++w) {
            if (redGV[w] > bGV || (redGV[w] == bGV && redGI[w] < bGI)) { bGV = redGV[w]; bGI = redGI[w]; }
            if (redSV[w] > bSV || (redSV[w] == bSV && redSI[w] < bSI)) { bSV = redSV[w]; bSI = redSI[w]; }
        }
        partial[blockIdx.x] = make_float4(bGV, __int_as_float(bGI),
                                          bSV, __int_as_float(bSI));
    }
}

// ---------------- stage 2: merge 5 segments per row, select token ----------------
__global__ __launch_bounds__(BLOCK) void sampler_finalize_kernel(
    const float4* __restrict__ partial,
    const float* __restrict__ temps,
    int* __restrict__ out, int B)
{
    const int r = blockIdx.x * BLOCK + threadIdx.x;
    if (r >= B) return;

    float bGV = -3.4e38f; int bGI = 0;
    float bSV = -3.4e38f; int bSI = 0;
#pragma unroll
    for (int s = 0; s < SEGS; ++s) {            // in segment order -> first-idx ties
        const float4 p = partial[r * SEGS + s];
        const float gv = p.x; const int gi = __float_as_int(p.y);
        const float sv = p.z; const int si = __float_as_int(p.w);
        if (gv > bGV || (gv == bGV && gi < bGI)) { bGV = gv; bGI = gi; }
        if (sv > bSV || (sv == bSV && si < bSI)) { bSV = sv; bSI = si; }
    }
    out[r] = (temps[r] == 0.0f) ? bGI : bSI;
}

extern "C" void kernel_launch(void* const* d_in, const int* in_sizes, int n_in,
                              void* d_out, int out_size, void* d_ws, size_t ws_size,
                              hipStream_t stream) {
    (void)n_in; (void)out_size; (void)ws_size;
    const float* logits = (const float*)d_in[0];
    const float* temps  = (const float*)d_in[1];
    int*    out     = (int*)d_out;
    float4* partial = (float4*)d_ws;            // B*SEGS*16 B = 20 KB scratch
    const int B = in_sizes[1];                  // 256 rows

    sampler_partial_kernel<<<B * SEGS, BLOCK, 0, stream>>>(logits, temps, partial);
    sampler_finalize_kernel<<<(B + BLOCK - 1) / BLOCK, BLOCK, 0, stream>>>(
        partial, temps, out, B);
}